// TopK_53369263620480
// MI455X (gfx1250) — compile-verified
//
#include <hip/hip_runtime.h>

// ---------------- problem geometry ----------------
#define ROWLEN   32768
#define TPB      1024
#define NWAVES   (TPB / 32)            // 32 waves (wave32)
#define NCHUNK   8
#define CHUNK    (ROWLEN / NCHUNK)     // 4096 elements per chunk
#define ELT      (ROWLEN / TPB)        // 32 elements per thread
#define EPC      (CHUNK / TPB)         // 4 elements per thread per chunk (16B = one b128)
#define KSEL     64

#define HISTPAD  257                   // 257-word stride => adjacent LDS banks across wave copies
// smem layout (uint words), phases separated by barriers:
//   Phase A (load):  stage0[4096] | stage1[4096]                  = 8192 words
//   Phase B (select): whist[32][257]=8224 | tot[256]@8224 | misc@8480
#define SMEM_WORDS 8512

typedef unsigned int u32x4 __attribute__((ext_vector_type(4)));
typedef float        f32x4 __attribute__((ext_vector_type(4)));

// ---------------- CDNA5 async global->LDS path (signature probe-confirmed) ----------------
#if defined(__HIP_DEVICE_COMPILE__) && defined(__gfx1250__) && \
    __has_builtin(__builtin_amdgcn_global_load_async_to_lds_b128)
#define USE_ASYNC 1
#else
#define USE_ASYNC 0
#endif

#if defined(__HIP_DEVICE_COMPILE__) && defined(__gfx1250__)
#if __has_builtin(__builtin_amdgcn_s_wait_asynccnt)
#define WAIT_ASYNC(n) __builtin_amdgcn_s_wait_asynccnt(n)
#else
#define WAIT_ASYNC(n) asm volatile("s_wait_asynccnt %0" ::"n"(n) : "memory")
#endif
#if __has_builtin(__builtin_amdgcn_s_wait_dscnt)
#define WAIT_DS0() __builtin_amdgcn_s_wait_dscnt(0)
#else
#define WAIT_DS0() asm volatile("s_wait_dscnt 0" ::: "memory")
#endif
#else
#define WAIT_ASYNC(n) do {} while (0)
#define WAIT_DS0()    do {} while (0)
#endif

#if USE_ASYNC
// exact pointee type from the compiler diagnostic: vector_size(16) int
typedef int v4i __attribute__((vector_size(16)));
typedef __attribute__((address_space(1))) v4i* gv4_p;  // prints as "int4 __device__ *"
typedef __attribute__((address_space(3))) v4i* lv4_p;  // prints as "int4 __shared__ *"
// generic->AS casts via integer: global flat == AS1 numerically; flat-shared low 32 bits
// ARE the LDS offset (ISA 10.2 aperture truncation)
__device__ __forceinline__ gv4_p gbl_cast(const void* p) {
    return (gv4_p)(unsigned long long)p;
}
__device__ __forceinline__ lv4_p lds_cast(void* p) {
    return (lv4_p)(unsigned int)(unsigned long long)p;
}
#define ASYNC_B128(gp, lp) \
    __builtin_amdgcn_global_load_async_to_lds_b128(gbl_cast(gp), lds_cast(lp), 0, 0)
#endif

// monotone key transform: larger float <-> larger unsigned key
__device__ __forceinline__ unsigned key_of(unsigned u) {
    return u ^ ((unsigned)((int)u >> 31) | 0x80000000u);
}
__device__ __forceinline__ float float_of_key(unsigned k) {
    unsigned u = (k & 0x80000000u) ? (k ^ 0x80000000u) : ~k;
    return __uint_as_float(u);
}

__global__ __launch_bounds__(TPB) void topk_rowselect(const float* __restrict__ z,
                                                      float* __restrict__ out) {
    __shared__ __align__(16) unsigned int smem[SMEM_WORDS];
    const int tid  = threadIdx.x;
    const int lane = tid & 31;
    const int wid  = tid >> 5;
    const long long rowbase = (long long)blockIdx.x * (long long)ROWLEN;
    const float* zrow = z + rowbase;

    unsigned int key[ELT];

    // ---------- Phase A: stream row into registers ----------
#if USE_ASYNC
    {
        unsigned int* stage0 = smem;
        unsigned int* stage1 = smem + CHUNK;
        // prime the double buffer (each thread moves its own 16B per chunk -> no cross-thread hazard)
        ASYNC_B128(zrow + 0 * CHUNK + tid * EPC, stage0 + tid * EPC);
        ASYNC_B128(zrow + 1 * CHUNK + tid * EPC, stage1 + tid * EPC);
#pragma unroll
        for (int c = 0; c < NCHUNK - 2; ++c) {
            WAIT_ASYNC(1);  // chunk c landed (async loads complete in order)
            asm volatile("" ::: "memory");
            unsigned int* slot = (c & 1) ? stage1 : stage0;
            u32x4 v = *(const u32x4*)(slot + tid * EPC);
            key[c * 4 + 0] = v.x; key[c * 4 + 1] = v.y;
            key[c * 4 + 2] = v.z; key[c * 4 + 3] = v.w;
            WAIT_DS0();     // our ds_load consumed the slot before the async engine overwrites it
            ASYNC_B128(zrow + (c + 2) * CHUNK + tid * EPC, slot + tid * EPC);
        }
        WAIT_ASYNC(1);
        asm volatile("" ::: "memory");
        {
            u32x4 v = *(const u32x4*)(stage0 + tid * EPC);  // chunk 6 (even slot)
            key[24] = v.x; key[25] = v.y; key[26] = v.z; key[27] = v.w;
        }
        WAIT_ASYNC(0);
        asm volatile("" ::: "memory");
        {
            u32x4 v = *(const u32x4*)(stage1 + tid * EPC);  // chunk 7 (odd slot)
            key[28] = v.x; key[29] = v.y; key[30] = v.z; key[31] = v.w;
        }
    }
#else
#pragma unroll
    for (int c = 0; c < NCHUNK; ++c) {
        u32x4 v = __builtin_nontemporal_load((const u32x4*)(zrow + c * CHUNK + tid * EPC));
        key[c * 4 + 0] = v.x; key[c * 4 + 1] = v.y;
        key[c * 4 + 2] = v.z; key[c * 4 + 3] = v.w;
    }
#endif

#pragma unroll
    for (int i = 0; i < ELT; ++i) key[i] = key_of(key[i]);

    __syncthreads();  // staging buffers free -> reuse LDS for histograms

    // ---------- Phase B: radix-256 select of the K-th largest key ----------
    unsigned int* whist = smem;                          // [NWAVES][HISTPAD]
    unsigned int* tot   = smem + NWAVES * HISTPAD;       // [256]
    unsigned int* misc  = smem + NWAVES * HISTPAD + 256; // [0]=prefix,[1]=want,[2]=tie tickets

    unsigned prefix = 0;
    unsigned want   = KSEL;

#pragma unroll
    for (int pass = 3; pass >= 0; --pass) {
        const int shift = pass * 8;
        const unsigned pmask  = (pass == 3) ? 0u : (0xFFFFFFFFu << (shift + 8));
        const unsigned pfx_hi = prefix & pmask;

        unsigned int* h = whist + wid * HISTPAD;  // this wave's private copy
        for (int b = lane; b < 256; b += 32) h[b] = 0;
        // wave-private: zero/add need no cross-wave sync; ds ops are in-order per wave

#pragma unroll
        for (int i = 0; i < ELT; ++i) {
            unsigned k = key[i];
            if ((k & pmask) == pfx_hi) atomicAdd(&h[(k >> shift) & 255u], 1u);
        }
        __syncthreads();

        if (tid < 256) {  // fold the 32 wave copies
            unsigned s = 0;
#pragma unroll
            for (int w = 0; w < NWAVES; ++w) s += whist[w * HISTPAD + tid];
            tot[tid] = s;
        }
        __syncthreads();

        if (tid < 32) {   // single-wave suffix scan over 256 bins (8 bins/lane)
            unsigned loc[8], s = 0;
#pragma unroll
            for (int j = 0; j < 8; ++j) { loc[j] = tot[lane * 8 + j]; s += loc[j]; }
            unsigned incl = s;
#pragma unroll
            for (int off = 1; off < 32; off <<= 1) {
                unsigned o = __shfl_down(incl, off, 32);
                incl += (lane + off < 32) ? o : 0u;
            }
            unsigned excl = incl - s;          // count of keys in strictly higher bins
            if (excl < want && incl >= want) { // exactly one lane holds the crossing
                unsigned run = excl;
#pragma unroll
                for (int j = 7; j >= 0; --j) {
                    unsigned nrun = run + loc[j];
                    if (nrun >= want) {
                        misc[0] = prefix | ((unsigned)(lane * 8 + j) << shift);
                        misc[1] = want - run;  // rank remaining inside selected bin
                        break;
                    }
                    run = nrun;
                }
            }
        }
        __syncthreads();
        prefix = misc[0];
        want   = misc[1];
        __syncthreads();
    }

    const unsigned thresh = prefix;  // exact key of the K-th largest element
    if (tid == 0) misc[2] = 0;
    __syncthreads();

    // ---------- Phase C: scatter relu'd top-K, zeros elsewhere ----------
    float* orow = out + rowbase;
#pragma unroll
    for (int c = 0; c < NCHUNK; ++c) {
        f32x4 o;
#pragma unroll
        for (int j = 0; j < 4; ++j) {
            unsigned k = key[c * 4 + j];
            float v = 0.0f;
            bool keep = (k > thresh);
            if (k == thresh) {  // ration boundary ties (>=1 element; usually exactly 1)
                unsigned t = atomicAdd(&misc[2], 1u);
                keep = (t < want);
            }
            if (keep) {
                float f = float_of_key(k);
                v = f > 0.0f ? f : 0.0f;  // RELU_AFTER
            }
            o[j] = v;
        }
        __builtin_nontemporal_store(o, (f32x4*)(orow + c * CHUNK + tid * EPC));
    }
}

extern "C" void kernel_launch(void* const* d_in, const int* in_sizes, int n_in,
                              void* d_out, int out_size, void* d_ws, size_t ws_size,
                              hipStream_t stream) {
    (void)n_in; (void)out_size; (void)d_ws; (void)ws_size;
    const float* z = (const float*)d_in[0];
    float* out = (float*)d_out;
    const int rows = in_sizes[0] / ROWLEN;  // 4096
    topk_rowselect<<<dim3(rows), dim3(TPB), 0, stream>>>(z, out);
}